// AllModel_66907000537726
// MI455X (gfx1250) — compile-verified
//
#include <hip/hip_runtime.h>
#include <math.h>

typedef __attribute__((ext_vector_type(2))) float v2f;
typedef __attribute__((ext_vector_type(8))) float v8f;

#define N_NODES 4000
#define N_EDGES 4000
#define T_TOT   8000
#define D_IN    20
#define HID     768
#define MP_E    4032   /* 4000 padded up to multiple of 64 */
#define KP1     128    /* 120 padded up to multiple of 32 */
#define KP2     32     /* 20 padded */

/* ---- workspace layout (units of 4-byte words) ---- */
#define WS_BITMAP   0            /* 500000 u32 : 16M-key reverse-edge bitmap */
#define WS_SEL      500000       /* 8000 int   : interleave gather indices   */
#define WS_ACCUM    508000       /* 4 float    : loss accumulators           */
#define WS_W1P      508004       /* 128*768  = 98304  */
#define WS_W2P      606308       /* 32*768   = 24576  */
#define WS_DATAP    630884       /* 4032*32  = 129024 */
#define WS_ERAW     759908       /* 4032*128 = 516096 */
#define WS_EH       1276004      /* 4032*768 = 3096576 (also reused for n_h) */
#define WS_EFEAT    4372580      /* 4032*768 */
#define WS_NFEAT    7469156      /* 4032*768 */
#define WS_X        10565732     /* 8000*768 = 6144000 */
#define WS_XW       WS_EFEAT     /* reuse efeat+nfeat (6193152 >= 6144000) */
#define WS_HW       WS_ERAW      /* reuse eraw (16000 <= 516096) */

/* ------------------------------------------------------------------ */
__global__ __launch_bounds__(256) void zero_u32_kernel(unsigned* p, int n) {
  int i = blockIdx.x * 256 + threadIdx.x;
  if (i < n) p[i] = 0u;
}

__global__ __launch_bounds__(256) void set_bits_kernel(const int* __restrict__ trans,
                                                       unsigned* __restrict__ bitmap) {
  int e = blockIdx.x * 256 + threadIdx.x;
  if (e < N_EDGES) {
    int t0 = trans[e], t1 = trans[N_EDGES + e];
    unsigned key = (unsigned)t0 * N_NODES + (unsigned)t1;
    atomicOr(&bitmap[key >> 5], 1u << (key & 31u));
  }
}

/* Build padded raw edge features [MP_E x 128]; cols 120..127 and rows >=4000 zero. */
__global__ __launch_bounds__(128) void build_eraw_kernel(const float* __restrict__ data,
                                                         const int* __restrict__ trans,
                                                         const unsigned* __restrict__ bitmap,
                                                         float* __restrict__ eraw) {
  int e = blockIdx.x;
  int c = threadIdx.x;          /* 0..127 */
  float val = 0.0f;
  if (e < N_EDGES) {
    int t0 = trans[e], t1 = trans[N_EDGES + e];
    unsigned rkey = (unsigned)t1 * N_NODES + (unsigned)t0;
    bool rev = (bitmap[rkey >> 5] >> (rkey & 31u)) & 1u;
    if (c < 20)      val = data[t1 * D_IN + c];
    else if (c < 40) val = data[t0 * D_IN + c - 20];
    else if (c < 60) val = rev ? data[t0 * D_IN + c - 40] : 0.0f;
    else if (c < 80) val = rev ? data[t1 * D_IN + c - 60] : 0.0f;
    else if (c < 120) {
      int cc = c - 80;
      float e1 = (cc < 20) ? data[t1 * D_IN + cc] : data[t0 * D_IN + cc - 20];
      float e2 = rev ? ((cc < 20) ? data[t0 * D_IN + cc] : data[t1 * D_IN + cc - 20]) : 0.0f;
      val = e1 + e2;
    }
  }
  eraw[e * KP1 + c] = val;
}

/* dst[R x C] = zero-padded copy of src[srcR x srcC] */
__global__ __launch_bounds__(256) void pad_copy_kernel(float* __restrict__ dst,
                                                       const float* __restrict__ src,
                                                       int R, int C, int srcR, int srcC) {
  int i = blockIdx.x * 256 + threadIdx.x;
  if (i < R * C) {
    int r = i / C, c = i % C;
    dst[i] = (r < srcR && c < srcC) ? src[r * srcC + c] : 0.0f;
  }
}

/* Sequential scan producing interleave gather indices (bit30 set => node row). */
__global__ void scan_sel_kernel(const int* __restrict__ inx, int* __restrict__ sel, int n) {
  if (blockIdx.x == 0 && threadIdx.x == 0) {
    int ce = 0, cn = 0;
    for (int i = 0; i < n; ++i) {
      if (inx[i] == 1) sel[i] = ce++;
      else             sel[i] = (cn++) | (1 << 30);
    }
  }
}

__global__ __launch_bounds__(256) void gather_x_kernel(const float* __restrict__ efeat,
                                                       const float* __restrict__ nfeat,
                                                       const int* __restrict__ sel,
                                                       float* __restrict__ x) {
  int i = blockIdx.x;                          /* 0..7999 */
  int c = blockIdx.y * 256 + threadIdx.x;      /* 0..767  */
  int s = sel[i];
  float v = (s & (1 << 30)) ? nfeat[(size_t)(s & ~(1 << 30)) * HID + c]
                            : efeat[(size_t)s * HID + c];
  x[(size_t)i * HID + c] = v;
}

/* ------------------------------------------------------------------ */
/* Generic f32 WMMA GEMM:  C[M,N] = op(A[M,K] @ B[K,N] + bias)         */
/* M % 64 == 0, N % 256 == 0, K % 32 == 0. 256 threads = 8 waves.      */
/* Block tile 64x256; wave w owns rows (w&1)*32, cols (w>>1)*64,       */
/* i.e. 2 A-fragments x 4 B-fragments -> 8 WMMAs per 6 ds_load_b64.    */
/* LDS layouts keep each fragment's (k,k+1) pair contiguous so every   */
/* fragment is a single aligned ds_load_b64 into the operand pair:     */
/*   sA: row-major, stride 34 floats (even => b64-aligned, 34*ll mod   */
/*       64 distinct banks for lanes 0-15).                            */
/*   sB: rows interleaved in pairs: float2 elem (k>>1, n), pair-stride */
/*       272 (2*272 mod 64 = 32 => high-half lanes on disjoint banks). */
#define TM 64
#define TN 256
#define TK 32
#define ASTR 34    /* floats per A row in LDS   */
#define BSTR 272   /* float2 per B pair-row     */

__global__ __launch_bounds__(256) void gemm_f32_wmma_kernel(
    const float* __restrict__ A, int lda,
    const float* __restrict__ B, int ldb,
    float* __restrict__ C, int ldc,
    int K, const float* __restrict__ bias, int do_relu) {
  __shared__ float sA[TM * ASTR];          /*  8704 B */
  __shared__ v2f   sB[(TK / 2) * BSTR];    /* 34816 B */
  const int m0  = blockIdx.x * TM;
  const int n0  = blockIdx.y * TN;
  const int tid = threadIdx.x;
  const int wave = tid >> 5;
  const int lane = tid & 31;
  const int wm = (wave & 1) * 32;   /* wave's row offset inside tile */
  const int wn = (wave >> 1) * 64;  /* wave's col offset inside tile */
  const int hl = lane >> 4;         /* 0 = low half, 1 = high half   */
  const int ll = lane & 15;

  v8f acc[2][4] = {};

  for (int k0 = 0; k0 < K; k0 += TK) {
    __syncthreads();
    /* A tile: 64x32 (2048 elems, 8/thread), coalesced along K */
    for (int e = tid; e < TM * TK; e += 256) {
      int r = e >> 5, c = e & 31;
      sA[r * ASTR + c] = A[(size_t)(m0 + r) * lda + (k0 + c)];
    }
    /* B tile: 32x256 as 16 pair-rows x 256 float2 (16/thread) */
    for (int e = tid; e < (TK / 2) * TN; e += 256) {
      int pr = e >> 8, c = e & 255;
      const float* bp = &B[(size_t)(k0 + 2 * pr) * ldb + (n0 + c)];
      v2f t;
      t.x = bp[0];
      t.y = bp[ldb];
      sB[pr * BSTR + c] = t;
    }
    __syncthreads();
#pragma unroll
    for (int kk = 0; kk < TK; kk += 4) {
      /* A frags 16x4: lanes0-15 -> K={kk,kk+1}; lanes16-31 -> K={kk+2,kk+3} */
      v2f a0 = *(const v2f*)&sA[(wm + ll) * ASTR + kk + 2 * hl];
      v2f a1 = *(const v2f*)&sA[(wm + 16 + ll) * ASTR + kk + 2 * hl];
      /* B frag 4x16: row br=kk+2*hl -> pair-row (kk>>1)+hl, (x,y)=(br,br+1) */
      const int pb = ((kk >> 1) + hl) * BSTR + wn + ll;
#pragma unroll
      for (int j = 0; j < 4; ++j) {
        v2f b = sB[pb + j * 16];
        acc[0][j] = __builtin_amdgcn_wmma_f32_16x16x4_f32(
            false, a0, false, b, (short)0, acc[0][j], false, false);
        acc[1][j] = __builtin_amdgcn_wmma_f32_16x16x4_f32(
            false, a1, false, b, (short)0, acc[1][j], false, false);
      }
    }
  }

  /* Epilogue. C/D layout: vgpr v, lanes0-15 -> M=v, lanes16-31 -> M=v+8 */
  const int rb0 = m0 + wm + 8 * hl;
#pragma unroll
  for (int r = 0; r < 2; ++r) {
#pragma unroll
    for (int j = 0; j < 4; ++j) {
      int col = n0 + wn + j * 16 + ll;
      float bv = bias ? bias[col] : 0.0f;
#pragma unroll
      for (int v = 0; v < 8; ++v) {
        float val = acc[r][j][v] + bv;
        if (do_relu) val = fmaxf(val, 0.0f);
        C[(size_t)(rb0 + 16 * r + v) * ldc + col] = val;
      }
    }
  }
}

/* ------------------------------------------------------------------ */
/* hw[i,{0,1}] = sum_k h1[i,k] * gw2[k,{0,1}]   (K = 768)             */
__global__ __launch_bounds__(256) void hw_kernel(const float* __restrict__ h1,
                                                 const float* __restrict__ gw2,
                                                 float* __restrict__ hw) {
  __shared__ float r0[256], r1[256];
  int i = blockIdx.x, t = threadIdx.x;
  float s0 = 0.f, s1 = 0.f;
  for (int k = t; k < HID; k += 256) {
    float h = h1[(size_t)i * HID + k];
    s0 += h * gw2[k * 2 + 0];
    s1 += h * gw2[k * 2 + 1];
  }
  r0[t] = s0; r1[t] = s1; __syncthreads();
  for (int off = 128; off > 0; off >>= 1) {
    if (t < off) { r0[t] += r0[t + off]; r1[t] += r1[t + off]; }
    __syncthreads();
  }
  if (t == 0) { hw[i * 2 + 0] = r0[0]; hw[i * 2 + 1] = r1[0]; }
}

/* logits[i,{0,1}] = sum_k A[i,k]*hw[k,{0,1}] + gb2  (K = 8000)        */
__global__ __launch_bounds__(256) void logits_kernel(const float* __restrict__ Aadj,
                                                     const float* __restrict__ hw,
                                                     const float* __restrict__ gb2,
                                                     float* __restrict__ logits) {
  __shared__ float r0[256], r1[256];
  int i = blockIdx.x, t = threadIdx.x;
  const float* row = Aadj + (size_t)i * T_TOT;
  float s0 = 0.f, s1 = 0.f;
  for (int k = t; k < T_TOT; k += 256) {
    float a = row[k];
    s0 += a * hw[k * 2 + 0];
    s1 += a * hw[k * 2 + 1];
  }
  r0[t] = s0; r1[t] = s1; __syncthreads();
  for (int off = 128; off > 0; off >>= 1) {
    if (t < off) { r0[t] += r0[t + off]; r1[t] += r1[t + off]; }
    __syncthreads();
  }
  if (t == 0) {
    logits[i * 2 + 0] = r0[0] + gb2[0];
    logits[i * 2 + 1] = r1[0] + gb2[1];
  }
}

__global__ __launch_bounds__(256) void loss_kernel(const float* __restrict__ logits,
                                                   const int* __restrict__ label,
                                                   const float* __restrict__ mask,
                                                   float* __restrict__ accum) {
  __shared__ float r0[256], r1[256];
  int t = threadIdx.x;
  int i = blockIdx.x * 256 + t;
  float v0 = 0.f, v1 = 0.f;
  if (i < T_TOT) {
    float l0 = logits[2 * i], l1 = logits[2 * i + 1];
    float mx = fmaxf(l0, l1);
    float lse = mx + logf(expf(l0 - mx) + expf(l1 - mx));
    float lp = (label[i] == 0 ? l0 : l1) - lse;
    float mk = mask[i];
    v0 = -lp * mk;
    v1 = mk;
  }
  r0[t] = v0; r1[t] = v1; __syncthreads();
  for (int off = 128; off > 0; off >>= 1) {
    if (t < off) { r0[t] += r0[t + off]; r1[t] += r1[t + off]; }
    __syncthreads();
  }
  if (t == 0) {
    atomicAdd(&accum[0], r0[0]);
    atomicAdd(&accum[1], r1[0]);
  }
}

__global__ void finalize_kernel(const float* __restrict__ accum, float* __restrict__ loss) {
  if (blockIdx.x == 0 && threadIdx.x == 0)
    loss[0] = accum[0] / fmaxf(accum[1], 1.0f);
}

/* ------------------------------------------------------------------ */
extern "C" void kernel_launch(void* const* d_in, const int* in_sizes, int n_in,
                              void* d_out, int out_size, void* d_ws, size_t ws_size,
                              hipStream_t stream) {
  (void)in_sizes; (void)n_in; (void)out_size; (void)ws_size;
  const float* data  = (const float*)d_in[0];
  /* d_in[1] = data_mask (unused by forward) */
  const float* Aadj  = (const float*)d_in[2];
  const int*   trans = (const int*)d_in[3];
  const int*   lip   = (const int*)d_in[4];
  const int*   label = (const int*)d_in[5];
  const float* mask  = (const float*)d_in[6];
  const float* w1    = (const float*)d_in[7];
  const float* b1    = (const float*)d_in[8];
  const float* w2    = (const float*)d_in[9];
  const float* b2    = (const float*)d_in[10];
  const float* m2w1  = (const float*)d_in[11];
  const float* m2b1  = (const float*)d_in[12];
  const float* m2w2  = (const float*)d_in[13];
  const float* m2b2  = (const float*)d_in[14];
  const float* gw1   = (const float*)d_in[15];
  const float* gb1   = (const float*)d_in[16];
  const float* gw2   = (const float*)d_in[17];
  const float* gb2   = (const float*)d_in[18];

  float*    ws     = (float*)d_ws;
  unsigned* bitmap = (unsigned*)d_ws;
  int*      sel    = (int*)(ws + WS_SEL);
  float*    accum  = ws + WS_ACCUM;
  float*    w1p    = ws + WS_W1P;
  float*    w2p    = ws + WS_W2P;
  float*    dataP  = ws + WS_DATAP;
  float*    eraw   = ws + WS_ERAW;
  float*    eh     = ws + WS_EH;
  float*    efeat  = ws + WS_EFEAT;
  float*    nh     = ws + WS_EH;     /* reuse: e_h dead after GEMM2 */
  float*    nfeat  = ws + WS_NFEAT;
  float*    x      = ws + WS_X;
  float*    xw     = ws + WS_XW;     /* reuse efeat+nfeat region    */
  float*    hw     = ws + WS_HW;     /* reuse eraw region           */

  float* logits = (float*)d_out;          /* [8000,2]       */
  float* loss   = logits + 16000;         /* scalar         */
  float* h1     = logits + 16001;         /* [8000,768]     */

  /* --- gen_edge2 --- */
  zero_u32_kernel<<<(500000 + 255) / 256, 256, 0, stream>>>(bitmap, 500000);
  zero_u32_kernel<<<1, 256, 0, stream>>>((unsigned*)accum, 4);
  set_bits_kernel<<<(N_EDGES + 255) / 256, 256, 0, stream>>>(trans, bitmap);
  build_eraw_kernel<<<MP_E, 128, 0, stream>>>(data, trans, bitmap, eraw);

  /* --- padded weight / data copies --- */
  pad_copy_kernel<<<(KP1 * HID + 255) / 256, 256, 0, stream>>>(w1p, w1, KP1, HID, 120, HID);
  pad_copy_kernel<<<(KP2 * HID + 255) / 256, 256, 0, stream>>>(w2p, m2w1, KP2, HID, D_IN, HID);
  pad_copy_kernel<<<(MP_E * KP2 + 255) / 256, 256, 0, stream>>>(dataP, data, MP_E, KP2, N_NODES, D_IN);

  /* --- interleave index scan --- */
  scan_sel_kernel<<<1, 32, 0, stream>>>(lip, sel, T_TOT);

  /* --- edge MLP: eh = relu(eraw @ w1p + b1); efeat = eh @ w2 + b2 --- */
  gemm_f32_wmma_kernel<<<dim3(MP_E / TM, HID / TN), 256, 0, stream>>>(
      eraw, KP1, w1p, HID, eh, HID, KP1, b1, 1);
  gemm_f32_wmma_kernel<<<dim3(MP_E / TM, HID / TN), 256, 0, stream>>>(
      eh, HID, w2, HID, efeat, HID, HID, b2, 0);

  /* --- node MLP: nh = relu(dataP @ w2p + m2b1); nfeat = nh @ m2w2 + m2b2 --- */
  gemm_f32_wmma_kernel<<<dim3(MP_E / TM, HID / TN), 256, 0, stream>>>(
      dataP, KP2, w2p, HID, nh, HID, KP2, m2b1, 1);
  gemm_f32_wmma_kernel<<<dim3(MP_E / TM, HID / TN), 256, 0, stream>>>(
      nh, HID, m2w2, HID, nfeat, HID, HID, m2b2, 0);

  /* --- interleave rows into x [8000,768] --- */
  gather_x_kernel<<<dim3(T_TOT, HID / 256), 256, 0, stream>>>(efeat, nfeat, sel, x);

  /* --- GCN layer 1: xw = x @ gw1 ; h1 = relu(A @ xw + gb1) --- */
  gemm_f32_wmma_kernel<<<dim3(T_TOT / TM, HID / TN), 256, 0, stream>>>(
      x, HID, gw1, HID, xw, HID, HID, nullptr, 0);
  gemm_f32_wmma_kernel<<<dim3(T_TOT / TM, HID / TN), 256, 0, stream>>>(
      Aadj, T_TOT, xw, HID, h1, HID, T_TOT, gb1, 1);

  /* --- GCN layer 2 (OUT=2): hw = h1 @ gw2 ; logits = A @ hw + gb2 --- */
  hw_kernel<<<T_TOT, 256, 0, stream>>>(h1, gw2, hw);
  logits_kernel<<<T_TOT, 256, 0, stream>>>(Aadj, hw, gb2, logits);

  /* --- masked cross-entropy loss --- */
  loss_kernel<<<(T_TOT + 255) / 256, 256, 0, stream>>>(logits, label, mask, accum);
  finalize_kernel<<<1, 32, 0, stream>>>(accum, loss);
}